// DetectionLoss_58909771432713
// MI455X (gfx1250) — compile-verified
//
#include <hip/hip_runtime.h>
#include <hip/hip_bf16.h>

#define GRID_N   13
#define NA       5
#define NC       20
#define CH       25
#define BATCH    2048
#define NOBJ     32
#define CELLS    (GRID_N * GRID_N)        /* 169 */
#define PLANE    (CH * CELLS)             /* 4225 */
#define NPLANES  (BATCH * NA)             /* 10240 */
#define NSLOT    (NPLANES * CELLS)        /* 1730560 = 6760 * 256 */
#define NGT      (BATCH * NOBJ)           /* 65536 */
#define NBLK     (NSLOT / 256)            /* 6760 */

typedef float v2f __attribute__((ext_vector_type(2)));
typedef float v8f __attribute__((ext_vector_type(8)));

static constexpr float ANCH[10] = {
    1.3221f, 1.73145f, 3.19275f, 4.00944f, 5.05587f,
    8.09892f, 9.47112f, 4.84053f, 11.2364f, 10.0071f};

// ---------------------------------------------------------------------------
// Deterministic block reduction of 4 accumulators across 256 threads using
// V_WMMA_F32_16X16X4_F32 (A = all-ones, so every row of D equals the column
// sums of B; the result is invariant to the exact A/B lane layout).
// ---------------------------------------------------------------------------
__device__ __forceinline__ void block_reduce4(float v0, float v1, float v2, float v3,
                                              float* __restrict__ out,
                                              float* s_P /* 256*4 floats */) {
  const int tid = threadIdx.x;
  s_P[tid * 4 + 0] = v0;
  s_P[tid * 4 + 1] = v1;
  s_P[tid * 4 + 2] = v2;
  s_P[tid * 4 + 3] = v3;
  __syncthreads();
  if (tid < 32) {
#if __has_builtin(__builtin_amdgcn_wmma_f32_16x16x4_f32)
    const int nn     = tid & 15;        // column n of B this lane feeds
    const int kb     = (tid >> 4) * 2;  // which K rows this lane supplies
    const int grp    = (nn >> 2) * 4;   // slot-group for this column
    const int accsel = nn & 3;          // accumulator index for this column
    v2f A;  A.x = 1.0f; A.y = 1.0f;     // ones(16x4)
    v8f C = {0.f, 0.f, 0.f, 0.f, 0.f, 0.f, 0.f, 0.f};
#pragma unroll
    for (int chunk = 0; chunk < 16; ++chunk) {
      const int base = chunk * 16 + grp;
      v2f Bv;
      Bv.x = s_P[(base + kb + 0) * 4 + accsel];
      Bv.y = s_P[(base + kb + 1) * 4 + accsel];
      C = __builtin_amdgcn_wmma_f32_16x16x4_f32(
          /*neg_a=*/false, A, /*neg_b=*/false, Bv,
          /*c_mod=*/(short)0, C, /*reuse_a=*/false, /*reuse_b=*/false);
    }
    // Every row of D is the column-sum vector; row M=0, N=lane sits in C[0]
    // for lanes 0..15.  Column n holds acc (n&3) over slot-group (n>>2).
    float cs = C[0];
    cs += __shfl_xor(cs, 4, 32);
    cs += __shfl_xor(cs, 8, 32);
    if (tid < 4) out[tid] = cs;   // [loss, obj, no_obj, conf]
#else
    if (tid == 0) {
      float s0 = 0.f, s1 = 0.f, s2 = 0.f, s3 = 0.f;
      for (int i = 0; i < 256; ++i) {
        s0 += s_P[i * 4 + 0];
        s1 += s_P[i * 4 + 1];
        s2 += s_P[i * 4 + 2];
        s3 += s_P[i * 4 + 3];
      }
      out[0] = s0; out[1] = s1; out[2] = s2; out[3] = s3;
    }
#endif
  }
}

// ---------------------------------------------------------------------------
__global__ void k_init(int* __restrict__ owner) {
  int i = blockIdx.x * blockDim.x + threadIdx.x;
  if (i < NSLOT) owner[i] = -1;
}

// One thread per ground-truth box: argmax-IoU anchor, scatter-max owner id.
__global__ void k_assign(const float* __restrict__ det,
                         const float* __restrict__ gtb,
                         int* __restrict__ owner) {
  int n = blockIdx.x * blockDim.x + threadIdx.x;
  if (n >= NGT) return;
  const int b = n >> 5;  // NOBJ == 32
  const float x = gtb[n * 4 + 0], y = gtb[n * 4 + 1];
  const float w = gtb[n * 4 + 2], h = gtb[n * 4 + 3];
  const int gx = (int)floorf(x * 13.0f);
  const int gy = (int)floorf(y * 13.0f);
  const int cell = gy * GRID_N + gx;
  const float gx0 = x - w * 0.5f, gy0 = y - h * 0.5f;
  const float gx1 = x + w * 0.5f, gy1 = y + h * 0.5f;
  const float a1 = (gx1 - gx0 + 1.f) * (gy1 - gy0 + 1.f);
  float best = -3.4e38f;
  int aidx = 0;
#pragma unroll
  for (int a = 0; a < NA; ++a) {
    const float* p = det + (size_t)(b * NA + a) * PLANE + cell;
    const float px = p[0 * CELLS];
    const float py = p[1 * CELLS];
    const float pw = p[2 * CELLS] * (ANCH[2 * a + 0] / 13.0f);
    const float ph = p[3 * CELLS] * (ANCH[2 * a + 1] / 13.0f);
    const float bx0 = (px + (float)gx) / 13.0f - pw * 0.5f;
    const float by0 = (py + (float)gy) / 13.0f - ph * 0.5f;
    const float bx1 = bx0 + pw, by1 = by0 + ph;
    const float ix0 = fmaxf(gx0, bx0), iy0 = fmaxf(gy0, by0);
    const float ix1 = fminf(gx1, bx1), iy1 = fminf(gy1, by1);
    const float inter = (ix1 - ix0 + 1.f) * (iy1 - iy0 + 1.f);
    const float a2 = (bx1 - bx0 + 1.f) * (by1 - by0 + 1.f);
    const float iou = inter / (a1 + a2 - inter);
    if (iou > best) { best = iou; aidx = a; }
  }
  atomicMax(owner + (size_t)(b * NA + aidx) * CELLS + cell, n);
}

// ---------------------------------------------------------------------------
// One thread per slot.  Invalid slots (weight==0 except conf channel) only
// touch owner + channel-4: terms with weight 0 are exactly 0 and are skipped,
// cutting HBM traffic from 173 MB to ~tens of MB.  Valid slots (<=65536) do
// the full 25-channel weighted-MSE with compile-time-constant byte offsets.
// ---------------------------------------------------------------------------
__global__ void k_loss(const float* __restrict__ det,
                       const float* __restrict__ gtb,
                       const int* __restrict__ gtc,
                       const int* __restrict__ owner,
                       float* __restrict__ partials) {
  __shared__ float s_P[256 * 4];
  const int s     = blockIdx.x * 256 + threadIdx.x;   // slot id, < NSLOT
  const int plane = s / CELLS;
  const int r     = s - plane * CELLS;                // cell in plane
  const int a     = plane % NA;
  const float* base = det + (size_t)plane * PLANE + r;
  const int   n  = owner[s];
  const float v4 = base[4 * CELLS];                   // conf channel, everyone

  float lsum = 0.f, osum = 0.f, nsum = 0.f, csum = 0.f;
  if (n < 0) {
    const float t = 0.5f * v4 * v4;                   // wt 0.5, gt 0
    lsum = t;
    nsum = t;
  } else {
    const float aw = ANCH[2 * a + 0] / 13.0f;
    const float ah = ANCH[2 * a + 1] / 13.0f;
    const float x = gtb[n * 4 + 0], y = gtb[n * 4 + 1];
    const float w = gtb[n * 4 + 2], h = gtb[n * 4 + 3];
    const int cls = gtc[n];
    const float tx = x * 13.0f - (float)(r % GRID_N);
    const float ty = y * 13.0f - (float)(r / GRID_N);
    const float d0 = tx - base[0 * CELLS];
    const float d1 = ty - base[1 * CELLS];
    const float d2 = w - base[2 * CELLS] * aw;
    const float d3 = h - base[3 * CELLS] * ah;
    const float d4 = 1.0f - v4;
    osum = 5.0f * (d0 * d0) + 5.0f * (d1 * d1) +
           5.0f * (d2 * d2) + 5.0f * (d3 * d3) + d4 * d4;
    float cacc = 0.f;
#pragma unroll
    for (int c = 0; c < NC; ++c) {
      const float g  = (c == cls) ? 1.0f : 0.0f;
      const float dd = g - base[(5 + c) * CELLS];
      cacc += dd * dd;
    }
    csum = cacc;
    lsum = osum + cacc;   // loss == obj + conf contributions for valid slots
  }
  block_reduce4(lsum, osum, nsum, csum, partials + (size_t)blockIdx.x * 4, s_P);
}

__global__ void k_final(const float* __restrict__ partials,
                        float* __restrict__ out) {
  __shared__ float s_P[256 * 4];
  const int tid = threadIdx.x;
  float s0 = 0.f, s1 = 0.f, s2 = 0.f, s3 = 0.f;
  for (int r = tid; r < NBLK; r += 256) {
    s0 += partials[r * 4 + 0];
    s1 += partials[r * 4 + 1];
    s2 += partials[r * 4 + 2];
    s3 += partials[r * 4 + 3];
  }
  block_reduce4(s0, s1, s2, s3, out, s_P);
}

// ---------------------------------------------------------------------------
extern "C" void kernel_launch(void* const* d_in, const int* in_sizes, int n_in,
                              void* d_out, int out_size, void* d_ws, size_t ws_size,
                              hipStream_t stream) {
  const float* det = (const float*)d_in[0];
  const float* gtb = (const float*)d_in[1];
  const int*   gtc = (const int*)d_in[2];
  float* out = (float*)d_out;

  int*   owner    = (int*)d_ws;
  float* partials = (float*)((char*)d_ws + (size_t)NSLOT * sizeof(int));

  k_init  <<<(NSLOT + 255) / 256, 256, 0, stream>>>(owner);
  k_assign<<<(NGT   + 255) / 256, 256, 0, stream>>>(det, gtb, owner);
  k_loss  <<<NBLK,                256, 0, stream>>>(det, gtb, gtc, owner, partials);
  k_final <<<1,                   256, 0, stream>>>(partials, out);
}